// ChamferDistance_68307159875939
// MI455X (gfx1250) — compile-verified
//
#include <hip/hip_runtime.h>
#include <hip/hip_bf16.h>

typedef __attribute__((ext_vector_type(2))) float v2f;
typedef __attribute__((ext_vector_type(4))) float v4f;
typedef __attribute__((ext_vector_type(8))) float v8f;
typedef __attribute__((ext_vector_type(4))) int   v4i;

typedef __attribute__((address_space(1))) v4i gv4i;   // global (AS1) int4
typedef __attribute__((address_space(3))) v4i lv4i;   // LDS    (AS3) int4

#define WAVES_PER_BLOCK 8
#define BLOCK_THREADS   (WAVES_PER_BLOCK * 32)
#define CHUNK_TILES     512                      // 512 tiles * 256B = 128 KB LDS
#define TILE_BYTES      256                      // 32 lanes * 2 floats * 4B
#define CHUNK_BYTES     (CHUNK_TILES * TILE_BYTES)

// ---------------------------------------------------------------------------
// Prep: swizzle both point sets into WMMA-ready layouts.
//   A-form (DB role):   per 16-pt tile t, lane l<16 holds (x,y) of point t*16+l,
//                       lane l+16 holds (z, |p|^2)  -> 16x4 f32 A layout
//   B-form (query role): lane l<16 -> (-2x,-2y) = rows K=0,1 of column l
//                        lane l+16 -> (-2z, 1 ) = rows K=2,3 of column l
//   S: |q|^2 per point.
// ---------------------------------------------------------------------------
__global__ void chamfer_prep(const float* __restrict__ xyz1,
                             const float* __restrict__ xyz2,
                             float* __restrict__ A1, float* __restrict__ A2,
                             float* __restrict__ B1, float* __restrict__ B2,
                             float* __restrict__ S1, float* __restrict__ S2,
                             int B, int N, int M)
{
    int u = blockIdx.x * blockDim.x + threadIdx.x;
    int total = B * (N + M);
    if (u >= total) return;

    int b, p, P;
    const float* src;
    float *A, *Bf, *S;
    if (u < B * N) { b = u / N; p = u % N; P = N; src = xyz1; A = A1; Bf = B1; S = S1; }
    else { int v = u - B * N; b = v / M; p = v % M; P = M; src = xyz2; A = A2; Bf = B2; S = S2; }

    size_t sp = (size_t)(b * P + p) * 3;
    float x = src[sp + 0], y = src[sp + 1], z = src[sp + 2];
    float sq = x * x + y * y + z * z;

    int t = p >> 4, r = p & 15;
    size_t lo = (size_t)b * 4 * P + ((size_t)t * 32 + r) * 2;
    size_t hi = lo + 32;

    A[lo] = x;          A[lo + 1] = y;
    A[hi] = z;          A[hi + 1] = sq;
    Bf[lo] = -2.0f * x; Bf[lo + 1] = -2.0f * y;
    Bf[hi] = -2.0f * z; Bf[hi + 1] = 1.0f;
    S[(size_t)b * P + p] = sq;
}

// ---------------------------------------------------------------------------
// Main: block = 8 waves = 8 query tiles of one (batch, direction).
// DB set staged into LDS in 128 KB chunks via async global->LDS; hot loop is
// ds_load_b64 + v_wmma_f32_16x16x4_f32 + 8 in-lane min/argmin updates.
// ---------------------------------------------------------------------------
__global__ __launch_bounds__(BLOCK_THREADS, 1)
void chamfer_nn(const float* __restrict__ A1, const float* __restrict__ A2,
                const float* __restrict__ B1, const float* __restrict__ B2,
                const float* __restrict__ S1, const float* __restrict__ S2,
                float* __restrict__ out, int B, int N, int M)
{
    extern __shared__ float smem[];

    int lane   = threadIdx.x & 31;
    int waveId = threadIdx.x >> 5;

    int qt1 = N >> 4, qt2 = M >> 4;          // query tiles per direction
    int bl1 = qt1 / WAVES_PER_BLOCK, bl2 = qt2 / WAVES_PER_BLOCK;
    int perB = bl1 + bl2;

    int blk = blockIdx.x;
    int b   = blk / perB;
    int rem = blk % perB;
    int dir = (rem < bl1) ? 0 : 1;           // 0: queries=xyz1, db=xyz2
    int bg  = dir ? (rem - bl1) : rem;
    int tq  = bg * WAVES_PER_BLOCK + waveId;

    int Qn = dir ? M : N;                    // query-set size
    int Dn = dir ? N : M;                    // db-set size
    int qtiles = dir ? qt2 : qt1;
    if (tq >= qtiles) tq = qtiles - 1;       // uniform clamp (no barrier skew)

    const float* Bq = dir ? B2 : B1;
    const float* Ap = dir ? A1 : A2;
    const float* Sq = dir ? S2 : S1;
    int dtiles = Dn >> 4;

    // B operand: constant for the whole kernel
    const v2f* bptr = (const v2f*)(Bq + (size_t)b * 4 * Qn);
    v2f bv = bptr[tq * 32 + lane];

    float best  = 3.402823466e38f;
    int   bidx  = 0;
    int   hioff = (lane >> 4) << 3;          // rows 0..7 vs 8..15 of each tile

    const char* gA = (const char*)(Ap + (size_t)b * 4 * Dn);
    int nChunks = (dtiles + CHUNK_TILES - 1) / CHUNK_TILES;

    for (int ch = 0; ch < nChunks; ++ch) {
        int tbase    = ch * CHUNK_TILES;
        int curTiles = dtiles - tbase; if (curTiles > CHUNK_TILES) curTiles = CHUNK_TILES;
        int curBytes = curTiles * TILE_BYTES;

        // ---- cooperative stage: global -> LDS (async DMA path) ----
        const char* gsrc = gA + (size_t)tbase * TILE_BYTES;
        char*       ldst = (char*)smem;
        for (int off = threadIdx.x * 16; off < curBytes; off += BLOCK_THREADS * 16) {
#if __has_builtin(__builtin_amdgcn_global_load_async_to_lds_b128)
            __builtin_amdgcn_global_load_async_to_lds_b128(
                (gv4i*)(gsrc + off), (lv4i*)(ldst + off), 0, 0);
#else
            *(v4f*)(ldst + off) = *(const v4f*)(gsrc + off);
#endif
        }
#if __has_builtin(__builtin_amdgcn_s_wait_asynccnt)
        __builtin_amdgcn_s_wait_asynccnt(0);
#else
        asm volatile("s_wait_asynccnt 0x0" ::: "memory");
#endif
        __syncthreads();

        // ---- hot loop: LDS tile -> WMMA -> argmin ----
        const v2f* lbase = (const v2f*)smem;
#pragma unroll 4
        for (int t = 0; t < curTiles; ++t) {
            v2f a = lbase[t * 32 + lane];    // ds_load_b64, bank-conflict-free
            v8f c = {};
            // D[m][n] = |db_m|^2 - 2*db_m . q_n   (K=4 packing)
            c = __builtin_amdgcn_wmma_f32_16x16x4_f32(
                    false, a, false, bv, (short)0, c, false, false);

            int ib = (tbase + t) * 16 + hioff;
#pragma unroll
            for (int r = 0; r < 8; ++r) {
                float d  = c[r];
                int   id = ib + r;
                bool  lt = d < best;         // strict < keeps first occurrence
                best = lt ? d : best;
                bidx = lt ? id : bidx;
            }
        }
        __syncthreads();                      // all waves done before restage
    }

    // Combine the two lane halves (rows 0..7+16k vs 8..15+16k of same column)
    float oD = __shfl_xor(best, 16, 32);
    int   oI = __shfl_xor(bidx, 16, 32);
    bool take = (oD < best) || (oD == best && oI < bidx);
    best = take ? oD : best;
    bidx = take ? oI : bidx;

    int q = tq * 16 + (lane & 15);
    best += Sq[(size_t)b * Qn + q];           // add |q|^2

    if (lane < 16) {
        // out layout: dist1[B*N] | dist2[B*M] | idx1[B*N] | idx2[B*M]
        size_t distOff = dir ? ((size_t)B * N + (size_t)b * M + q)
                             : ((size_t)b * N + q);
        size_t idxOff  = (size_t)B * (N + M) + distOff;
        out[distOff] = best;
        ((int*)out)[idxOff] = bidx;
    }
}

extern "C" void kernel_launch(void* const* d_in, const int* in_sizes, int n_in,
                              void* d_out, int out_size, void* d_ws, size_t ws_size,
                              hipStream_t stream)
{
    const int B = 2;
    const int N = in_sizes[0] / (3 * B);
    const int M = in_sizes[1] / (3 * B);

    const float* xyz1 = (const float*)d_in[0];
    const float* xyz2 = (const float*)d_in[1];
    float* ws = (float*)d_ws;

    size_t oA1 = 0;
    size_t oA2 = oA1 + (size_t)B * 4 * N;
    size_t oB1 = oA2 + (size_t)B * 4 * M;
    size_t oB2 = oB1 + (size_t)B * 4 * N;
    size_t oS1 = oB2 + (size_t)B * 4 * M;
    size_t oS2 = oS1 + (size_t)B * N;

    float *A1 = ws + oA1, *A2 = ws + oA2;
    float *B1 = ws + oB1, *B2 = ws + oB2;
    float *S1 = ws + oS1, *S2 = ws + oS2;

    int prepThreads = B * (N + M);
    chamfer_prep<<<(prepThreads + 255) / 256, 256, 0, stream>>>(
        xyz1, xyz2, A1, A2, B1, B2, S1, S2, B, N, M);

    // allow 128 KB dynamic LDS
    (void)hipFuncSetAttribute((const void*)chamfer_nn,
                              hipFuncAttributeMaxDynamicSharedMemorySize,
                              CHUNK_BYTES);

    int blocks = B * ((N >> 4) / WAVES_PER_BLOCK + (M >> 4) / WAVES_PER_BLOCK);
    chamfer_nn<<<blocks, BLOCK_THREADS, CHUNK_BYTES, stream>>>(
        A1, A2, B1, B2, S1, S2, (float*)d_out, B, N, M);
}